// CARAFE_51488067945157
// MI455X (gfx1250) — compile-verified
//
#include <hip/hip_runtime.h>
#include <stdint.h>

// CARAFE (N=4, C=128, H=W=64, k=5, scale=2, group=1) for gfx1250.
// Core math on the matrix pipe: V_WMMA_F32_16X16X4_F32 (full fp32 precision).
// Per wave-tile: D(16ch x 16px) = A(16ch x 36 footprint taps) * B(36 x 16px),
// where 16px = 2x2 coarse block x 2x2 subpixels sharing one 6x6 feature
// footprint -> 9 wmma per tile at ~69% utilization, B reused over 8 ch-tiles.
// Features staged global->LDS with async-to-LDS (transpose for free).

#define B_N   4
#define B_C   128
#define B_H   64
#define B_W   64
#define UPH   128
#define UPW   128
#define KER   5
#define KSQ   25

typedef __attribute__((ext_vector_type(2))) float v2f;
typedef __attribute__((ext_vector_type(8))) float v8f;

typedef __attribute__((address_space(1))) int as1_int;
typedef __attribute__((address_space(3))) int as3_int;

#define LDSPITCH 133   // pitch for [64 spatial][128 ch] tile; 133 coprime w/ 64 banks

__global__ __launch_bounds__(128) void carafe_wmma_f32(
    const float* __restrict__ feat,   // (4,128,64,64)
    const float* __restrict__ mask,   // (4,25,128,128)
    float* __restrict__ out)          // (4,128,128,128)
{
  __shared__ float ldsF[64 * LDSPITCH];   // ~33 KB of 320 KB

  const int bid = blockIdx.x;             // 1024 blocks: 4 imgs x 16x16 regions
  const int n   = bid >> 8;
  const int rid = bid & 255;
  const int h0  = (rid >> 4) << 2;        // 4x4 coarse region origin
  const int w0  = (rid & 15) << 2;

  const int tid = threadIdx.x;            // 128 threads = 4 waves

  // ---- stage features: rows h0-2..h0+5, cols w0-2..w0+5, all 128 ch ----
  // LDS layout (transposed): ldsF[spatial * LDSPITCH + channel]
  {
    const int xl = tid & 7;
    const int yl = (tid >> 3) & 7;
    const int cl = tid >> 6;              // 0..1
    const int si = yl * 8 + xl;
    const int gy = h0 - 2 + yl;
    const int gx = w0 - 2 + xl;
    const bool inb = ((unsigned)gy < B_H) && ((unsigned)gx < B_W);
    for (int it = 0; it < 64; ++it) {
      const int c  = (it << 1) + cl;
      const int li = si * LDSPITCH + c;
      if (inb) {
        const float* gp = feat + (((size_t)n * B_C + c) * B_H + gy) * B_W + gx;
#if __has_builtin(__builtin_amdgcn_global_load_async_to_lds_b32)
        __builtin_amdgcn_global_load_async_to_lds_b32(
            (as1_int*)gp,
            (as3_int*)&ldsF[li],
            0, 0);
#else
        // NOTE: assumes ldsF (sole __shared__ object) sits at LDS offset 0.
        const unsigned loff = (unsigned)(li * 4);
        asm volatile("global_load_async_to_lds_b32 %0, %1, off"
                     :: "v"(loff), "v"((unsigned long long)(uintptr_t)gp)
                     : "memory");
#endif
      } else {
        ldsF[li] = 0.0f;                  // zero-pad border
      }
    }
  }
#if __has_builtin(__builtin_amdgcn_s_wait_asynccnt)
  __builtin_amdgcn_s_wait_asynccnt(0);
#else
  asm volatile("s_wait_asynccnt 0" ::: "memory");
#endif
  __syncthreads();

  // ---- per-wave tile setup ----
  const int lane = tid & 31;
  const int wv   = tid >> 5;
  const int dy2  = wv >> 1;               // wave's 2x2 coarse block in region
  const int dx2  = wv & 1;
  const int hb   = h0 + (dy2 << 1);       // coarse block origin
  const int wb   = w0 + (dx2 << 1);

  const int mrow = lane & 15;             // A: M (channel-local); D: N (pixel)
  const int kg   = lane >> 4;             // K-group: 0 -> K{0,1}, 1 -> K{2,3}

  // Column n = (uy,ux) in the 4x4 upsampled block of this coarse 2x2 block.
  const int uy = mrow >> 2;
  const int ux = mrow & 3;
  const int hu = (hb << 1) + uy;          // hu = 2*(hb+dy)+a with uy=2*dy+a
  const int wu = (wb << 1) + ux;
  const int dy = uy >> 1;                 // coarse offset of this column
  const int dx = ux >> 1;

  // ---- build B (36 x 16, 9 K-chunks) once; reuse for all 8 channel tiles ----
  // footprint tap k = r*6+s, r,s in [0,6); column uses mask tap (i,j)=(r-dy,s-dx)
  v2f Bv[9];
  int aoff0[9], aoff1[9];                 // LDS word offsets for A taps (per lane)
#pragma unroll
  for (int ch = 0; ch < 9; ++ch) {
    const int k0 = (ch << 2) + (kg << 1); // this lane's two K rows
    float b0 = 0.0f, b1 = 0.0f;
    {
      const int r = k0 / 6, s = k0 % 6;
      const int i = r - dy, j = s - dx;
      if ((unsigned)i < KER && (unsigned)j < KER)
        b0 = mask[(((size_t)n * KSQ + (i * KER + j)) * UPH + hu) * UPW + wu];
      aoff0[ch] = (((dy2 << 1) + r) * 8 + ((dx2 << 1) + s)) * LDSPITCH + mrow;
    }
    {
      const int k1 = k0 + 1;
      const int r = k1 / 6, s = k1 % 6;
      const int i = r - dy, j = s - dx;
      if ((unsigned)i < KER && (unsigned)j < KER)
        b1 = mask[(((size_t)n * KSQ + (i * KER + j)) * UPH + hu) * UPW + wu];
      aoff1[ch] = (((dy2 << 1) + r) * 8 + ((dx2 << 1) + s)) * LDSPITCH + mrow;
    }
    v2f b; b.x = b0; b.y = b1;
    Bv[ch] = b;
  }

  // ---- channel loop: 8 tiles of 16 channels, 9 wmma each ----
#pragma unroll 1
  for (int ct = 0; ct < 8; ++ct) {
    const int c0 = ct << 4;
    v8f acc = {};
#pragma unroll
    for (int ch = 0; ch < 9; ++ch) {
      v2f a;
      a.x = ldsF[aoff0[ch] + c0];         // A: lane ch-row = c0+mrow, taps k0,k0+1
      a.y = ldsF[aoff1[ch] + c0];
      acc = __builtin_amdgcn_wmma_f32_16x16x4_f32(
          /*neg_a=*/false, a, /*neg_b=*/false, Bv[ch],
          /*c_mod=*/(short)0, acc, /*reuse_a=*/false, /*reuse_b=*/false);
    }
    // D: lane holds column (hu,wu); VGPR v holds channel c0 + kg*8 + v
    const int cw = c0 + (kg << 3);
    float* op = out + (((size_t)n * B_C + cw) * UPH + hu) * UPW + wu;
#pragma unroll
    for (int v = 0; v < 8; ++v)
      op[(size_t)v * UPH * UPW] = acc[v];
  }
}

extern "C" void kernel_launch(void* const* d_in, const int* in_sizes, int n_in,
                              void* d_out, int out_size, void* d_ws, size_t ws_size,
                              hipStream_t stream) {
  const float* feat = (const float*)d_in[0];
  const float* mask = (const float*)d_in[1];
  float* out = (float*)d_out;
  (void)in_sizes; (void)n_in; (void)out_size; (void)d_ws; (void)ws_size;
  const int blocks = B_N * (B_H / 4) * (B_W / 4);   // 1024
  carafe_wmma_f32<<<blocks, 128, 0, stream>>>(feat, mask, out);
}